// MetaAttention_5093831213743
// MI455X (gfx1250) — compile-verified
//
#include <hip/hip_runtime.h>
#include <hip/hip_bf16.h>

// ---------------- problem constants ----------------
#define B_  16
#define H_  12
#define N_  784
#define TILE_ROWS 16            // i-rows per block (matches WMMA M)
#define LDS_STRIDE 788          // padded LDS row stride (floats): 16B aligned, bank-conflict free
#define NTILES 49               // N_/TILE_ROWS
#define CHUNKS 3136             // TILE_ROWS*(N_/4) b128 chunks per head tile
#define ISSUE_ITERS 13          // ceil(CHUNKS/256) async issues per thread per head
#define COLS_PER_THREAD 49      // TILE_ROWS*N_/256

typedef __attribute__((ext_vector_type(2))) float v2f;
typedef __attribute__((ext_vector_type(8))) float v8f;

typedef int v4i __attribute__((vector_size(16)));
typedef __attribute__((address_space(1))) v4i as1_v4i;
typedef __attribute__((address_space(3))) v4i as3_v4i;
typedef __attribute__((address_space(3))) void as3_void;

#if __has_builtin(__builtin_amdgcn_wmma_f32_16x16x4_f32)
#define USE_WMMA32 1
#else
#define USE_WMMA32 0
#endif

// ---- CDNA5 async global -> LDS (tracked by ASYNCcnt) ----
__device__ __forceinline__ void async_load_b128(const void* g, void* l) {
#if __has_builtin(__builtin_amdgcn_global_load_async_to_lds_b128)
  __builtin_amdgcn_global_load_async_to_lds_b128((as1_v4i*)g, (as3_v4i*)l, 0, 0);
#else
  unsigned lds_off = (unsigned)(unsigned long long)(as3_void*)l;
  asm volatile("global_load_async_to_lds_b128 %0, %1, off"
               :: "v"(lds_off), "v"(g) : "memory");
#endif
}

template <int Cnt>
__device__ __forceinline__ void wait_asynccnt() {
#if __has_builtin(__builtin_amdgcn_s_wait_asynccnt)
  __builtin_amdgcn_s_wait_asynccnt(Cnt);
#else
  asm volatile("s_wait_asynccnt %0" :: "i"(Cnt) : "memory");
#endif
}

// =====================================================================
// Kernel 0: init per-batch max (all aggregated values are >= 0)
// =====================================================================
__global__ void k_init(int* __restrict__ bmax_i) {
  if (threadIdx.x < B_) bmax_i[threadIdx.x] = 0;   // +0.0f bit pattern
}

// =====================================================================
// Kernel 1: stream masks, rowsum via f32 WMMA, aggregate over heads,
//           write agg into d_out, atomic-max per batch.
// =====================================================================
__global__ void __launch_bounds__(256, 1)
meta_main(const float* __restrict__ m, float* __restrict__ agg_out,
          int* __restrict__ bmax_i) {
  __shared__ float buf[2][TILE_ROWS][LDS_STRIDE];   // double-buffered head tile
  __shared__ float parts[8][16];                    // per-wave rowsum partials
  __shared__ float rsum[16];
  __shared__ float red[256];

  const int t     = threadIdx.x;
  const int b     = blockIdx.x / NTILES;
  const int itile = blockIdx.x - b * NTILES;
  const int i0    = itile * TILE_ROWS;

  const int lane  = t & 31;
  // wave id is uniform across the wave: make it scalar so the WMMA loop has
  // scalar control flow (EXEC structurally all-1s, as the ISA requires).
  const int wvs   = __builtin_amdgcn_readfirstlane(t >> 5);
  const int arow  = lane & 15;     // WMMA A-matrix row (M)
  const int ahalf = lane >> 4;     // 0: K=0,1  1: K=2,3

  const int crow = t >> 4;         // accumulate row (0..15), 16 threads/row
  const int ccol = t & 15;         // column phase: cols = ccol + 16*s

  float acc[COLS_PER_THREAD];
#pragma unroll
  for (int s = 0; s < COLS_PER_THREAD; ++s) acc[s] = 0.f;

  auto issue_head = [&](int h, int pb) {
    const float* srcbh = m + (((long)b * H_ + h) * N_ + i0) * (long)N_;
#pragma unroll
    for (int it = 0; it < ISSUE_ITERS; ++it) {
      int q = t + it * 256;
      q = (q < CHUNKS) ? q : (CHUNKS - 1);   // clamp: uniform issue count per wave
      int row = q / 196;
      int c4  = q - row * 196;
      async_load_b128(srcbh + row * N_ + c4 * 4, &buf[pb][row][c4 * 4]);
    }
  };

  issue_head(0, 0);
  issue_head(1, 1);
  wait_asynccnt<ISSUE_ITERS>();    // head 0 landed (in-order async completion)
  __syncthreads();

  for (int h = 0; h < H_; ++h) {
    const int pb = h & 1;

    // ---- rowsums of this head tile: WMMA f32 16x16x4 against all-ones B ----
#if USE_WMMA32
    {
      v8f cf = {0.f, 0.f, 0.f, 0.f, 0.f, 0.f, 0.f, 0.f};
      v2f bones; bones.x = 1.0f; bones.y = 1.0f;
#pragma unroll 2
      for (int c = wvs; c < 196; c += 8) {   // scalar bound: EXEC stays all-1s
        int koff = c * 4 + ahalf * 2;
        v2f a = *(const v2f*)&buf[pb][arow][koff];
        cf = __builtin_amdgcn_wmma_f32_16x16x4_f32(
            false, a, false, bones, (short)0, cf, false, false);
      }
      // C columns are identical: lane0 holds rows 0..7, lane16 holds rows 8..15
      if (lane == 0) {
#pragma unroll
        for (int v = 0; v < 8; ++v) parts[wvs][v] = cf[v];
      }
      if (lane == 16) {
#pragma unroll
        for (int v = 0; v < 8; ++v) parts[wvs][8 + v] = cf[v];
      }
    }
#else
    {
      float ps = 0.f;
      for (int c = wvs; c < 196; c += 8) {
        int koff = c * 4 + ahalf * 2;
        ps += buf[pb][arow][koff] + buf[pb][arow][koff + 1];
      }
      ps += __shfl_down(ps, 16);
      if (lane < 16) parts[wvs][lane] = ps;
    }
#endif
    __syncthreads();
    if (t < 16) {
      float s = 0.f;
#pragma unroll
      for (int w2 = 0; w2 < 8; ++w2) s += parts[w2][t];
      rsum[t] = s;
    }
    __syncthreads();

    // ---- aggregated[i,l] += rowsum[i] * m[h,i,l] ----
    {
      const float r = rsum[crow];
#pragma unroll
      for (int s = 0; s < COLS_PER_THREAD; ++s)
        acc[s] = fmaf(r, buf[pb][crow][ccol + s * 16], acc[s]);
    }

    if (h + 1 < H_) {
      __syncthreads();                    // everyone finished reading buf[pb]
      if (h + 2 < H_) {
        issue_head(h + 2, pb);            // prefetch into the buffer just freed
        wait_asynccnt<ISSUE_ITERS>();     // head h+1 landed, h+2 in flight
      } else {
        wait_asynccnt<0>();               // drain: head h+1 landed
      }
      __syncthreads();
    }
  }

  // ---- write aggregated (into d_out) + per-batch max ----
  float mx = 0.f;
  float* outp = agg_out + ((long)b * N_ + (i0 + crow)) * N_ + ccol;
#pragma unroll
  for (int s = 0; s < COLS_PER_THREAD; ++s) {
    float v = acc[s];
    mx = fmaxf(mx, v);
    outp[s * 16] = v;
  }
  red[t] = mx;
  __syncthreads();
#pragma unroll
  for (int off = 128; off > 0; off >>= 1) {
    if (t < off) red[t] = fmaxf(red[t], red[t + off]);
    __syncthreads();
  }
  if (t == 0) atomicMax(&bmax_i[b], __float_as_int(red[0]));  // exact, order-free
}

// =====================================================================
// Kernel 2: per (batch, slice) partial sum of exp(a - max), float4 loads.
// Slice = 9604 floats = 2401 aligned float4s. Deterministic fixed-order tree.
// =====================================================================
__global__ void __launch_bounds__(256)
k_sumexp(const float* __restrict__ agg, const int* __restrict__ bmax_i,
         float* __restrict__ partial) {
  __shared__ float red[256];
  const int b     = blockIdx.x >> 6;
  const int slice = blockIdx.x & 63;
  const int per4  = (N_ * N_) / 64 / 4;   // 2401 float4 per slice
  const float mx  = __int_as_float(bmax_i[b]);
  const float4* p = (const float4*)(agg + (long)b * N_ * N_) + (long)slice * per4;
  float s = 0.f;
  for (int i = threadIdx.x; i < per4; i += 256) {
    float4 a = p[i];
    s += __expf(a.x - mx) + __expf(a.y - mx) + __expf(a.z - mx) + __expf(a.w - mx);
  }
  red[threadIdx.x] = s;
  __syncthreads();
#pragma unroll
  for (int off = 128; off > 0; off >>= 1) {
    if (threadIdx.x < off) red[threadIdx.x] += red[threadIdx.x + off];
    __syncthreads();
  }
  if (threadIdx.x == 0) partial[blockIdx.x] = red[0];
}

// =====================================================================
// Kernel 3: reduce 64 partials per batch (fixed order -> deterministic)
// =====================================================================
__global__ void k_reduce64(const float* __restrict__ partial,
                           float* __restrict__ bsum) {
  __shared__ float red[64];
  red[threadIdx.x] = partial[blockIdx.x * 64 + threadIdx.x];
  __syncthreads();
#pragma unroll
  for (int off = 32; off > 0; off >>= 1) {
    if (threadIdx.x < off) red[threadIdx.x] += red[threadIdx.x + off];
    __syncthreads();
  }
  if (threadIdx.x == 0) bsum[blockIdx.x] = red[0];
}

// =====================================================================
// Kernel 4: out = exp(agg - max)/sum, in place, float4 vectorized.
// =====================================================================
__global__ void __launch_bounds__(256)
k_norm(float* __restrict__ out, const int* __restrict__ bmax_i,
       const float* __restrict__ bsum) {
  const long i4 = (long)blockIdx.x * 256 + threadIdx.x;        // float4 index
  const int  b  = (int)(i4 / ((long)(N_ * N_) / 4));
  const float mx  = __int_as_float(bmax_i[b]);
  const float inv = 1.f / bsum[b];
  float4 a = ((const float4*)out)[i4];
  float4 r;
  r.x = __expf(a.x - mx) * inv;
  r.y = __expf(a.y - mx) * inv;
  r.z = __expf(a.z - mx) * inv;
  r.w = __expf(a.w - mx) * inv;
  ((float4*)out)[i4] = r;
}

// =====================================================================
extern "C" void kernel_launch(void* const* d_in, const int* in_sizes, int n_in,
                              void* d_out, int out_size, void* d_ws, size_t ws_size,
                              hipStream_t stream) {
  (void)in_sizes; (void)n_in; (void)out_size; (void)ws_size;
  const float* m = (const float*)d_in[0];
  float* out = (float*)d_out;          // holds aggregated, then normalized in place
  float* wsf = (float*)d_ws;
  int*   bmax_i  = (int*)wsf;          // 16 ints (float bits, values >= 0)
  float* partial = wsf + 16;           // 16*64 floats
  float* bsum    = wsf + 16 + B_ * 64; // 16 floats

  k_init   <<<1, 32, 0, stream>>>(bmax_i);
  meta_main<<<B_ * NTILES, 256, 0, stream>>>(m, out, bmax_i);
  k_sumexp <<<B_ * 64, 256, 0, stream>>>(out, bmax_i, partial);
  k_reduce64<<<B_, 64, 0, stream>>>(partial, bsum);
  k_norm   <<<(B_ * N_ * N_ / 4) / 256, 256, 0, stream>>>(out, bmax_i, bsum);
}